// Model_49486613185097
// MI455X (gfx1250) — compile-verified
//
#include <hip/hip_runtime.h>
#include <hip/hip_bf16.h>
#include <math.h>

#define U_CNT 100000
#define I_CNT 50000
#define N_CNT 150000
#define D_DIM 64
#define NE_UI 1000000
#define NE_ENC 2000000
#define NE_DEC 1000000
#define IMG_DF 1024
#define TXT_DF 384

typedef float v2f __attribute__((ext_vector_type(2)));
typedef float v8f __attribute__((ext_vector_type(8)));

// ---------------------------------------------------------------------------
// Dense GEMM: C[M x 64] = A[M x K] @ W[K x 64] (+ optional bias)
// v_wmma_f32_16x16x4_f32; block = 4 waves; block -> 16 M-rows, wave w -> cols
// [16w,16w+16). A tile staged in LDS (stride 68 -> conflict-free), B streamed
// from L2-resident weights. Requires M % 16 == 0, K % 64 == 0 (true here).
// ---------------------------------------------------------------------------
__global__ __launch_bounds__(128) void gemm_n64_wmma(
    const float* __restrict__ A, const float* __restrict__ W,
    const float* __restrict__ bias, float* __restrict__ C, int M, int K) {
  __shared__ float lA[16 * 68];
  const int tid  = threadIdx.x;
  const int wave = tid >> 5;
  const int lane = tid & 31;
  const int t = lane & 15;   // M row (A) / N col (B,C)
  const int h = lane >> 4;   // K half selector
  const int m0 = blockIdx.x << 4;
  const int col = (wave << 4) + t;

  v8f acc = {};
  for (int k0 = 0; k0 < K; k0 += 64) {
    {  // cooperative A-tile load: 16 rows x 64 cols, 8 floats per thread
      int idx = tid << 3;
      int r = idx >> 6;
      int c = idx & 63;
      const float4* src = (const float4*)(A + (size_t)(m0 + r) * K + k0 + c);
      float4 p0 = src[0];
      float4 p1 = src[1];
      float* dst = &lA[r * 68 + c];
      ((float4*)dst)[0] = p0;
      ((float4*)dst)[1] = p1;
    }
    __syncthreads();
#pragma unroll
    for (int kk = 0; kk < 64; kk += 4) {
      v2f a, b;
      // A 16x4 layout: lane(t,h): vgpr0 = A[t][kk+2h], vgpr1 = A[t][kk+2h+1]
      a.x = lA[t * 68 + kk + 2 * h];
      a.y = lA[t * 68 + kk + 2 * h + 1];
      // B 4x16 layout mirrors A: vgpr0 = B[kk+2h][col], vgpr1 = B[kk+2h+1][col]
      const float* wp = W + (size_t)(k0 + kk + 2 * h) * 64 + col;
      b.x = wp[0];
      b.y = wp[64];
      acc = __builtin_amdgcn_wmma_f32_16x16x4_f32(
          false, a, false, b, (short)0, acc, false, false);
    }
    __syncthreads();
  }
  float bv = bias ? bias[col] : 0.0f;
#pragma unroll
  for (int r = 0; r < 8; ++r) {
    // C/D layout: vgpr r holds M = r (lanes 0-15) and M = r+8 (lanes 16-31)
    C[(size_t)(m0 + r + 8 * h) * 64 + col] = acc[r] + bv;
  }
}

// ---------------------------------------------------------------------------
// SpMM scatter: out[rows[e]] += vals[e] * x[cols[e]]  (out pre-zeroed)
// 64 contiguous threads per edge -> coalesced 256B gather + 256B atomics.
// ---------------------------------------------------------------------------
__global__ __launch_bounds__(256) void spmm_scatter(
    const int* __restrict__ rows, const int* __restrict__ cols,
    const float* __restrict__ vals, const float* __restrict__ x,
    float* __restrict__ out, int E) {
  size_t gid = (size_t)blockIdx.x * blockDim.x + threadIdx.x;
  size_t e = gid >> 6;
  int d = (int)(gid & 63);
  if (e >= (size_t)E) return;
  int r = rows[e];
  int c = cols[e];
  float v = vals[e];
  atomicAdd(out + (size_t)r * 64 + d, v * x[(size_t)c * 64 + d]);
}

// cl[i] = cur[i] = (i < U*64 ? uEmb[i] : iEmb[i - U*64])
__global__ __launch_bounds__(256) void concat_init(
    const float* __restrict__ uEmb, const float* __restrict__ iEmb,
    float* __restrict__ cl, float* __restrict__ cur) {
  size_t i = (size_t)blockIdx.x * blockDim.x + threadIdx.x;
  size_t nu = (size_t)U_CNT * 64;
  size_t nt = (size_t)N_CNT * 64;
  if (i >= nt) return;
  float v = (i < nu) ? uEmb[i] : iEmb[i - nu];
  cl[i] = v;
  cur[i] = v;
}

__global__ __launch_bounds__(256) void vec_add_inplace(
    float* __restrict__ dst, const float* __restrict__ src, size_t n) {
  size_t i = (size_t)blockIdx.x * blockDim.x + threadIdx.x;
  if (i < n) dst[i] += src[i];
}

// ---------------------------------------------------------------------------
// GT attention pass 1: per-edge per-head dot(q,k), clip, exp, accumulate
// softmax denominator. 64 threads/edge = 2 wave32; heads are 16-lane groups,
// reduced with a 4-step shfl_xor butterfly (stays inside each 16-lane group).
// ---------------------------------------------------------------------------
__global__ __launch_bounds__(256) void gt_att_exp(
    const int* __restrict__ rows, const int* __restrict__ cols,
    const float* __restrict__ Xq, const float* __restrict__ Xk,
    float* __restrict__ ex, float* __restrict__ attNorm, int E) {
  size_t gid = (size_t)blockIdx.x * blockDim.x + threadIdx.x;
  size_t e = gid >> 6;
  int d = (int)(gid & 63);
  if (e >= (size_t)E) return;
  int r = rows[e];
  int c = cols[e];
  float p = Xq[(size_t)r * 64 + d] * Xk[(size_t)c * 64 + d];
  p += __shfl_xor(p, 8, 32);
  p += __shfl_xor(p, 4, 32);
  p += __shfl_xor(p, 2, 32);
  p += __shfl_xor(p, 1, 32);
  if ((d & 15) == 0) {
    int hh = d >> 4;
    float att = fminf(fmaxf(p, -10.0f), 10.0f);
    float ev = expf(att);
    ex[e * 4 + hh] = ev;
    atomicAdd(attNorm + (size_t)r * 4 + hh, ev);
  }
}

// GT pass 2: out[rows[e]] += (ex/(norm+1e-8)) * Xv[cols[e]]   (out pre-zeroed)
__global__ __launch_bounds__(256) void gt_apply(
    const int* __restrict__ rows, const int* __restrict__ cols,
    const float* __restrict__ Xv, const float* __restrict__ ex,
    const float* __restrict__ attNorm, float* __restrict__ out, int E) {
  size_t gid = (size_t)blockIdx.x * blockDim.x + threadIdx.x;
  size_t e = gid >> 6;
  int d = (int)(gid & 63);
  if (e >= (size_t)E) return;
  int r = rows[e];
  int c = cols[e];
  int hh = d >> 4;
  float a = ex[e * 4 + hh] / (attNorm[(size_t)r * 4 + hh] + 1e-8f);
  atomicAdd(out + (size_t)r * 64 + d, a * Xv[(size_t)c * 64 + d]);
}

// ---------------------------------------------------------------------------
// Final combine: out = cl + 0.5*(l2n(a) + l2n(b)). One wave per 64-wide row;
// each lane owns dims (lane, lane+32); full-wave shfl_xor reduction.
// ---------------------------------------------------------------------------
__global__ __launch_bounds__(256) void l2n_combine(
    const float* __restrict__ cl, const float* __restrict__ a,
    const float* __restrict__ b, float* __restrict__ out, int nRows) {
  int wid = (int)(((size_t)blockIdx.x * blockDim.x + threadIdx.x) >> 5);
  int lane = threadIdx.x & 31;
  if (wid >= nRows) return;
  size_t base = (size_t)wid * 64;
  float a0 = a[base + lane], a1 = a[base + lane + 32];
  float b0 = b[base + lane], b1 = b[base + lane + 32];
  float sa = a0 * a0 + a1 * a1;
  float sb = b0 * b0 + b1 * b1;
#pragma unroll
  for (int m = 16; m; m >>= 1) {
    sa += __shfl_xor(sa, m, 32);
    sb += __shfl_xor(sb, m, 32);
  }
  float na = fmaxf(sqrtf(sa), 1e-12f);
  float nb = fmaxf(sqrtf(sb), 1e-12f);
  out[base + lane]      = cl[base + lane]      + 0.5f * (a0 / na + b0 / nb);
  out[base + lane + 32] = cl[base + lane + 32] + 0.5f * (a1 / na + b1 / nb);
}

// ---------------------------------------------------------------------------
extern "C" void kernel_launch(void* const* d_in, const int* in_sizes, int n_in,
                              void* d_out, int out_size, void* d_ws, size_t ws_size,
                              hipStream_t stream) {
  const float* uEmb     = (const float*)d_in[0];
  const float* iEmb     = (const float*)d_in[1];
  const float* imgFeats = (const float*)d_in[2];
  const float* txtFeats = (const float*)d_in[3];
  const float* img_W    = (const float*)d_in[4];
  const float* img_b    = (const float*)d_in[5];
  const float* txt_W    = (const float*)d_in[6];
  const float* txt_b    = (const float*)d_in[7];
  const float* ui_vals  = (const float*)d_in[8];
  const float* iu_vals  = (const float*)d_in[9];
  const float* enc_vals = (const float*)d_in[10];
  const float* qTrans   = (const float*)d_in[11];
  const float* kTrans   = (const float*)d_in[12];
  const float* vTrans   = (const float*)d_in[13];
  const int* ui_rows  = (const int*)d_in[14];
  const int* ui_cols  = (const int*)d_in[15];
  const int* iu_rows  = (const int*)d_in[16];
  const int* iu_cols  = (const int*)d_in[17];
  const int* enc_rows = (const int*)d_in[18];
  const int* enc_cols = (const int*)d_in[19];
  const int* dec_rows = (const int*)d_in[20];
  const int* dec_cols = (const int*)d_in[21];
  float* out = (float*)d_out;

  // ---- workspace carve (floats) ----
  float* ws = (float*)d_ws;
  size_t off = 0;
  auto carve = [&](size_t n) { float* p = ws + off; off += n; return p; };
  float* img_f = carve((size_t)I_CNT * 64);
  float* txt_f = carve((size_t)I_CNT * 64);
  float* img_u = carve((size_t)U_CNT * 64);
  float* txt_u = carve((size_t)U_CNT * 64);
  float* img_i = carve((size_t)I_CNT * 64);
  float* txt_i = carve((size_t)I_CNT * 64);
  float* cl    = carve((size_t)N_CNT * 64);
  float* cur   = carve((size_t)N_CNT * 64);
  float* nxt   = carve((size_t)N_CNT * 64);
  float* Xq    = carve((size_t)N_CNT * 64);
  float* Xk    = carve((size_t)N_CNT * 64);
  float* Xv    = carve((size_t)N_CNT * 64);
  float* exbuf = carve((size_t)NE_DEC * 4);
  float* attN  = carve((size_t)N_CNT * 4);
  (void)ws_size; (void)in_sizes; (void)n_in; (void)out_size;

  const size_t nvec = (size_t)N_CNT * 64;
  const unsigned addBlocks = (unsigned)((nvec + 255) / 256);

  // 1) modality feature projections (WMMA GEMMs)
  gemm_n64_wmma<<<I_CNT / 16, 128, 0, stream>>>(imgFeats, img_W, img_b, img_f, I_CNT, IMG_DF);
  gemm_n64_wmma<<<I_CNT / 16, 128, 0, stream>>>(txtFeats, txt_W, txt_b, txt_f, I_CNT, TXT_DF);

  // 2) modality graph propagation: item->user, then user->item
  hipMemsetAsync(img_u, 0, (size_t)U_CNT * 64 * sizeof(float), stream);
  hipMemsetAsync(txt_u, 0, (size_t)U_CNT * 64 * sizeof(float), stream);
  hipMemsetAsync(img_i, 0, (size_t)I_CNT * 64 * sizeof(float), stream);
  hipMemsetAsync(txt_i, 0, (size_t)I_CNT * 64 * sizeof(float), stream);
  {
    unsigned g = (unsigned)(((size_t)NE_UI * 64 + 255) / 256);
    spmm_scatter<<<g, 256, 0, stream>>>(ui_rows, ui_cols, ui_vals, img_f, img_u, NE_UI);
    spmm_scatter<<<g, 256, 0, stream>>>(ui_rows, ui_cols, ui_vals, txt_f, txt_u, NE_UI);
    spmm_scatter<<<g, 256, 0, stream>>>(iu_rows, iu_cols, iu_vals, img_u, img_i, NE_UI);
    spmm_scatter<<<g, 256, 0, stream>>>(iu_rows, iu_cols, iu_vals, txt_u, txt_i, NE_UI);
  }

  // 3) layer stack: x0 = concat(uEmb, iEmb); cl accumulates all layers
  concat_init<<<addBlocks, 256, 0, stream>>>(uEmb, iEmb, cl, cur);

  // 4) two GCN layers (SpMM over enc graph)
  for (int l = 0; l < 2; ++l) {
    hipMemsetAsync(nxt, 0, nvec * sizeof(float), stream);
    unsigned g = (unsigned)(((size_t)NE_ENC * 64 + 255) / 256);
    spmm_scatter<<<g, 256, 0, stream>>>(enc_rows, enc_cols, enc_vals, cur, nxt, NE_ENC);
    vec_add_inplace<<<addBlocks, 256, 0, stream>>>(cl, nxt, nvec);
    float* tmp = cur; cur = nxt; nxt = tmp;
  }

  // 5) two graph-transformer layers: project per NODE (gather commutes with @)
  for (int l = 0; l < 2; ++l) {
    const float* qT = qTrans + (size_t)l * 64 * 64;
    const float* kT = kTrans + (size_t)l * 64 * 64;
    const float* vT = vTrans + (size_t)l * 64 * 64;
    gemm_n64_wmma<<<N_CNT / 16, 128, 0, stream>>>(cur, qT, nullptr, Xq, N_CNT, 64);
    gemm_n64_wmma<<<N_CNT / 16, 128, 0, stream>>>(cur, kT, nullptr, Xk, N_CNT, 64);
    gemm_n64_wmma<<<N_CNT / 16, 128, 0, stream>>>(cur, vT, nullptr, Xv, N_CNT, 64);
    hipMemsetAsync(attN, 0, (size_t)N_CNT * 4 * sizeof(float), stream);
    hipMemsetAsync(nxt, 0, nvec * sizeof(float), stream);
    unsigned g = (unsigned)(((size_t)NE_DEC * 64 + 255) / 256);
    gt_att_exp<<<g, 256, 0, stream>>>(dec_rows, dec_cols, Xq, Xk, exbuf, attN, NE_DEC);
    gt_apply<<<g, 256, 0, stream>>>(dec_rows, dec_cols, Xv, exbuf, attN, nxt, NE_DEC);
    vec_add_inplace<<<addBlocks, 256, 0, stream>>>(cl, nxt, nvec);
    float* tmp = cur; cur = nxt; nxt = tmp;
  }

  // 6) final combine with l2-normalized modality embeddings
  l2n_combine<<<(U_CNT + 7) / 8, 256, 0, stream>>>(cl, img_u, txt_u, out, U_CNT);
  l2n_combine<<<(I_CNT + 7) / 8, 256, 0, stream>>>(
      cl + (size_t)U_CNT * 64, img_i, txt_i, out + (size_t)U_CNT * 64, I_CNT);
}